// mBoxFilter_70720931496761
// MI455X (gfx1250) — compile-verified
//
#include <hip/hip_runtime.h>
#include <stdint.h>

#define W        512
#define RAD      4
#define OUT_ROWS 32
#define TILE_ROWS (OUT_ROWS + 2 * RAD)   // 40
#define NBLK     (W / OUT_ROWS)          // 16 row-blocks per plane
#define NT       512                     // 16 wave32s

__global__ __launch_bounds__(NT) void box9_kernel(const float* __restrict__ x,
                                                  float* __restrict__ out) {
  // 40*512*4 = 80 KB raw + 80 KB horizontal-filtered = 160 KB (<= 320 KB/WGP, 2 WGs resident)
  __shared__ float raw [TILE_ROWS][W];
  __shared__ float hbuf[TILE_ROWS][W];

  const int tid   = threadIdx.x;
  const int plane = blockIdx.y;
  const int r0    = blockIdx.x * OUT_ROWS;

  // ---------------- Phase 0: async global -> LDS tile copy (CDNA5 path) ----
  // 40 rows x 512 cols = 5120 float4s; 512 threads * 10 iters, 16 B per lane.
  const unsigned long long gbase =
      (unsigned long long)(uintptr_t)(x + (size_t)plane * W * W);
#pragma unroll
  for (int k = 0; k < (TILE_ROWS * W / 4) / NT; ++k) {
    const int idx4 = tid + k * NT;
    const int tr   = idx4 >> 7;     // / (W/4)
    const int c4   = idx4 & 127;    // float4 index within row
    const int g    = r0 - RAD + tr; // global source row
    float* dst = &raw[tr][c4 * 4];
    if (g >= 0 && g < W) {
      const unsigned int ldsa = (unsigned int)(uintptr_t)dst;          // LDS byte addr (low 32b)
      const unsigned int gofs = (unsigned int)(g * (W * 4) + c4 * 16); // byte offset from base
      asm volatile("global_load_async_to_lds_b128 %0, %1, %2"
                   :: "v"(ldsa), "v"(gofs), "s"(gbase)
                   : "memory");
    } else {
      *(float4*)dst = make_float4(0.f, 0.f, 0.f, 0.f);  // zero padding rows
    }
  }
  asm volatile("s_wait_asynccnt 0" ::: "memory");  // ASYNCcnt drained -> LDS valid
  __syncthreads();

  // ---------------- Phase 1: horizontal 9-tap box sum (sliding window) -----
  const int lane = tid & 31;
  const int wv   = tid >> 5;
#pragma unroll 1
  for (int rr = wv; rr < TILE_ROWS; rr += NT / 32) {
    const float* rowp = raw[rr];
    // Need cols [16*lane-4 .. 16*lane+19] -> 6 aligned float4 blocks 4*lane-1 .. 4*lane+4
    float v[24];
#pragma unroll
    for (int bi = 0; bi < 6; ++bi) {
      const int blk = 4 * lane - 1 + bi;
      float4 q = (blk >= 0 && blk < W / 4) ? ((const float4*)rowp)[blk]
                                           : make_float4(0.f, 0.f, 0.f, 0.f);
      v[4 * bi + 0] = q.x; v[4 * bi + 1] = q.y;
      v[4 * bi + 2] = q.z; v[4 * bi + 3] = q.w;
    }
    float s = 0.f;
#pragma unroll
    for (int t = 0; t < 9; ++t) s += v[t];   // h at col 16*lane
    float h[16];
    h[0] = s;
#pragma unroll
    for (int j = 1; j < 16; ++j) {           // slide: +x[c+4] - x[c-5]
      s += v[j + 8] - v[j - 1];
      h[j] = s;
    }
    float4* hp = (float4*)&hbuf[rr][lane * 16];
    hp[0] = make_float4(h[0],  h[1],  h[2],  h[3]);
    hp[1] = make_float4(h[4],  h[5],  h[6],  h[7]);
    hp[2] = make_float4(h[8],  h[9],  h[10], h[11]);
    hp[3] = make_float4(h[12], h[13], h[14], h[15]);
  }
  __syncthreads();

  // ---------------- Phase 2: vertical 9-tap box sum, float4 per thread -----
  const int cg    = tid & 127;   // 4-column group
  const int chunk = tid >> 7;    // 0..3 -> 8 output rows each
  const int j     = cg * 4;
  const int base  = chunk * 8;   // local output row start; uses H rows base..base+15

  float4 s4 = make_float4(0.f, 0.f, 0.f, 0.f);
#pragma unroll
  for (int i = 0; i < 9; ++i) {
    const float4 a = *(const float4*)&hbuf[base + i][j];
    s4.x += a.x; s4.y += a.y; s4.z += a.z; s4.w += a.w;
  }
  float* op = out + ((size_t)plane * W + (size_t)(r0 + base)) * W + j;
  *(float4*)op = s4;
#pragma unroll
  for (int i = 1; i < 8; ++i) {
    const float4 a = *(const float4*)&hbuf[base + 8 + i][j];
    const float4 b = *(const float4*)&hbuf[base + i - 1][j];
    s4.x += a.x - b.x; s4.y += a.y - b.y;
    s4.z += a.z - b.z; s4.w += a.w - b.w;
    *(float4*)(op + (size_t)i * W) = s4;
  }
}

extern "C" void kernel_launch(void* const* d_in, const int* in_sizes, int n_in,
                              void* d_out, int out_size, void* d_ws, size_t ws_size,
                              hipStream_t stream) {
  const float* x = (const float*)d_in[0];
  float* out     = (float*)d_out;
  const int planes = in_sizes[0] / (W * W);   // 8*16 = 128
  dim3 grid(NBLK, planes, 1);
  box9_kernel<<<grid, NT, 0, stream>>>(x, out);
}